// DynamicVFE_13254269075962
// MI455X (gfx1250) — compile-verified
//
#include <hip/hip_runtime.h>

// ---------------- problem constants (from reference) ----------------
#define NPTS    400000
#define BATCH   2
#define GXc     352
#define GYc     400
#define GZc     1
#define CANVASc (BATCH*GZc*GYc*GXc)   // 281600
#define C0c     64
#define C1c     128
#define KPAD0   16                    // 10 input channels padded to 16
#define VXc     0.2f
#define VYc     0.2f
#define VZc     4.0f
#define PCR0f   0.0f
#define PCR1f   (-40.0f)
#define PCR2f   (-3.0f)
#define XOFFf   (VXc*0.5f + PCR0f)
#define YOFFf   (VYc*0.5f + PCR1f)
#define ZOFFf   (VZc*0.5f + PCR2f)
#define BNEPSf  1e-3f

typedef float v2f __attribute__((ext_vector_type(2)));
typedef float v8f __attribute__((ext_vector_type(8)));

// ---------------- CDNA5 async global->LDS copy (per-lane gather, ASYNCcnt) ----------------
__device__ __forceinline__ void async_g2lds_b32(unsigned lds_byte_off, const float* gptr) {
  // GLOBAL_LOAD_ASYNC_TO_LDS_B32: each lane copies one dword from its global address to its LDS byte offset
  asm volatile("global_load_async_to_lds_b32 %0, %1, off"
               :: "v"(lds_byte_off), "v"(gptr) : "memory");
}
__device__ __forceinline__ void wait_asynccnt0() {
  asm volatile("s_wait_asynccnt 0x0" ::: "memory");
}

// ---------------- helpers ----------------
__global__ void zero_f32(float* __restrict__ p, size_t n) {
  size_t i = (size_t)blockIdx.x * blockDim.x + threadIdx.x;
  size_t stride = (size_t)gridDim.x * blockDim.x;
  for (; i < n; i += stride) p[i] = 0.0f;
}

// zero-pad W0 (10x64) into W0p (16x64) so the GEMM0 inner loop has no branches
__global__ void pad_w0(const float* __restrict__ W0, float* __restrict__ W0p) {
  int i = blockIdx.x * blockDim.x + threadIdx.x;
  if (i >= KPAD0 * C0c) return;
  int k = i >> 6;
  W0p[i] = (k < 10) ? W0[i] : 0.0f;
}

// ---------------- pass 1: voxel scatter (count + sum xyz) ----------------
__global__ void voxelize(const float* __restrict__ pts, int* __restrict__ vidx,
                         float* __restrict__ cnt, float* __restrict__ sx,
                         float* __restrict__ sy, float* __restrict__ sz) {
  int p = blockIdx.x * blockDim.x + threadIdx.x;
  if (p >= NPTS) return;
  float b = pts[p*5+0], x = pts[p*5+1], y = pts[p*5+2], z = pts[p*5+3];
  int cx = min(max((int)floorf((x - PCR0f) / VXc), 0), GXc - 1);
  int cy = min(max((int)floorf((y - PCR1f) / VYc), 0), GYc - 1);
  int cz = min(max((int)floorf((z - PCR2f) / VZc), 0), GZc - 1);
  int v = (((int)b * GZc + cz) * GYc + cy) * GXc + cx;
  vidx[p] = v;
  atomicAdd(&cnt[v], 1.0f);
  atomicAdd(&sx[v], x);
  atomicAdd(&sy[v], y);
  atomicAdd(&sz[v], z);
}

// ---------------- pass 2: 10-channel point features (padded to 16) ----------------
__global__ void build_feats(const float* __restrict__ pts, const int* __restrict__ vidx,
                            const float* __restrict__ cnt, const float* __restrict__ sx,
                            const float* __restrict__ sy, const float* __restrict__ sz,
                            float* __restrict__ feats) {
  int p = blockIdx.x * blockDim.x + threadIdx.x;
  if (p >= NPTS) return;
  float x = pts[p*5+1], y = pts[p*5+2], z = pts[p*5+3], inten = pts[p*5+4];
  int v = vidx[p];
  float c = fmaxf(cnt[v], 1.0f);
  float mx = sx[v] / c, my = sy[v] / c, mz = sz[v] / c;
  int cx = min(max((int)floorf((x - PCR0f) / VXc), 0), GXc - 1);
  int cy = min(max((int)floorf((y - PCR1f) / VYc), 0), GYc - 1);
  int cz = min(max((int)floorf((z - PCR2f) / VZc), 0), GZc - 1);
  float* f = feats + (size_t)p * KPAD0;
  f[0] = x; f[1] = y; f[2] = z; f[3] = inten;
  f[4] = x - mx; f[5] = y - my; f[6] = z - mz;
  f[7] = x - ((float)cx * VXc + XOFFf);
  f[8] = y - ((float)cy * VYc + YOFFf);
  f[9] = z - ((float)cz * VZc + ZOFFf);
  f[10] = 0.0f; f[11] = 0.0f; f[12] = 0.0f; f[13] = 0.0f; f[14] = 0.0f; f[15] = 0.0f;
}

// ---------------- GEMM0: h0[N,64] = feats[N,16pad] @ W0p[16,64], + BN partials ----------------
// block: 256 thr = 8 waves; block tile = 32 points x 64 cols; wave tile = 16x16, K-steps of 4
#define LDS0_STRIDE 17
__global__ __launch_bounds__(256) void gemm0_wmma(const float* __restrict__ feats,
                                                  const float* __restrict__ W0p,
                                                  float* __restrict__ h0,
                                                  float* __restrict__ sum0,
                                                  float* __restrict__ sq0) {
  __shared__ float Atile[32 * LDS0_STRIDE];
  const int tid = threadIdx.x;
  const int blockRow = blockIdx.x * 32;           // NPTS % 32 == 0
  for (int i = tid; i < 32 * KPAD0; i += 256) {
    int r = i >> 4, c = i & 15;
    Atile[r * LDS0_STRIDE + c] = feats[(size_t)(blockRow + r) * KPAD0 + c];
  }
  __syncthreads();
  const int wave = tid >> 5, lane = tid & 31;
  const int mtile = wave >> 2, ntile = wave & 3;  // 2 M-tiles x 4 N-tiles
  const int l = lane & 15;
  const int khalf = (lane >> 4) << 1;             // 0 or 2
  const int col = ntile * 16 + l;
  v8f acc = {};
  for (int kk = 0; kk < 4; ++kk) {
    int k = kk * 4 + khalf;
    v2f a, b;
    a.x = Atile[(mtile * 16 + l) * LDS0_STRIDE + k];
    a.y = Atile[(mtile * 16 + l) * LDS0_STRIDE + k + 1];
    b.x = W0p[(k    ) * C0c + col];
    b.y = W0p[(k + 1) * C0c + col];
    acc = __builtin_amdgcn_wmma_f32_16x16x4_f32(false, a, false, b,
                                                (short)0, acc, false, false);
  }
  // D layout: VGPR r -> row r (lanes 0-15) / row r+8 (lanes 16-31)
  const int mbase = mtile * 16 + ((lane >> 4) << 3);
  float s = 0.0f, s2 = 0.0f;
  for (int r = 0; r < 8; ++r) {
    float vv = acc[r];
    h0[(size_t)(blockRow + mbase + r) * C0c + col] = vv;
    s += vv; s2 += vv * vv;
  }
  atomicAdd(&sum0[col], s);
  atomicAdd(&sq0[col], s2);
}

// ---------------- BN parameter fold: a = rsqrt(var+eps)*gamma, b' = beta - mu*a ----------------
__global__ void bn_params(const float* __restrict__ sum, const float* __restrict__ sq,
                          const float* __restrict__ gamma, const float* __restrict__ beta,
                          float* __restrict__ a, float* __restrict__ bshift, int C) {
  int c = blockIdx.x * blockDim.x + threadIdx.x;
  if (c >= C) return;
  float invN = 1.0f / (float)NPTS;
  float mu = sum[c] * invN;
  float var = sq[c] * invN - mu * mu;
  float aa = rsqrtf(var + BNEPSf) * gamma[c];
  a[c] = aa;
  bshift[c] = beta[c] - mu * aa;
}

// ---------------- fused BN + ReLU + segment-max (atomicMax on nonneg float bits) ----------------
__global__ void act_segmax(float* __restrict__ h, const float* __restrict__ a,
                           const float* __restrict__ bsh, const int* __restrict__ vidx,
                           float* __restrict__ canvas, int logC, int writeback) {
  const int C = 1 << logC;
  size_t total = (size_t)NPTS << logC;
  size_t i = (size_t)blockIdx.x * blockDim.x + threadIdx.x;
  size_t stride = (size_t)gridDim.x * blockDim.x;
  for (; i < total; i += stride) {
    int c = (int)(i & (size_t)(C - 1));
    size_t p = i >> logC;
    float v = fmaxf(fmaf(h[i], a[c], bsh[c]), 0.0f);   // relu(h*a + b')  >= 0
    if (writeback) h[i] = v;
    atomicMax((unsigned int*)&canvas[((size_t)vidx[p] << logC) + c],
              __float_as_uint(v));
  }
}

// ---------------- GEMM1: h1[N,128] = [pf0 | vmax0[vidx]] @ W1[128,128], + BN partials ----------------
// block: 256 thr = 8 waves; block tile = 16 points x 128 cols; wave = one 16x16 col tile; K=128
// A-tile fill uses CDNA5 GLOBAL_LOAD_ASYNC_TO_LDS_B32 (per-lane gather straight into LDS).
#define LDS1_STRIDE 132
__global__ __launch_bounds__(256) void gemm1_wmma(const float* __restrict__ pf0,
                                                  const float* __restrict__ vmax0,
                                                  const int* __restrict__ vidx,
                                                  const float* __restrict__ W1,
                                                  float* __restrict__ h1,
                                                  float* __restrict__ sum1,
                                                  float* __restrict__ sq1) {
  __shared__ float Atile[16 * LDS1_STRIDE];
  const int tid = threadIdx.x;
  const int blockRow = blockIdx.x * 16;           // NPTS % 16 == 0
  // async gather: cols 0-63 from pf0[p], cols 64-127 from vmax0[vidx[p]]
  for (int i = tid; i < 16 * C1c; i += 256) {
    int r = i >> 7, c = i & 127;
    int p = blockRow + r;
    const float* g = (c < 64) ? &pf0[(size_t)p * C0c + c]
                              : &vmax0[(size_t)vidx[p] * C0c + (c - 64)];
    unsigned lds_off = (unsigned)(uintptr_t)&Atile[r * LDS1_STRIDE + c];
    async_g2lds_b32(lds_off, g);
  }
  wait_asynccnt0();
  __syncthreads();
  const int wave = tid >> 5, lane = tid & 31;
  const int l = lane & 15;
  const int khalf = (lane >> 4) << 1;             // 0 or 2
  const int col = wave * 16 + l;                  // 8 waves cover 128 cols
  v8f acc = {};
  for (int kk = 0; kk < 32; ++kk) {
    int k = kk * 4 + khalf;
    v2f a, b;
    a.x = Atile[l * LDS1_STRIDE + k];
    a.y = Atile[l * LDS1_STRIDE + k + 1];
    b.x = W1[(size_t)(k    ) * C1c + col];
    b.y = W1[(size_t)(k + 1) * C1c + col];
    acc = __builtin_amdgcn_wmma_f32_16x16x4_f32(false, a, false, b,
                                                (short)0, acc, false, false);
  }
  const int mbase = (lane >> 4) << 3;
  float s = 0.0f, s2 = 0.0f;
  for (int r = 0; r < 8; ++r) {
    float vv = acc[r];
    h1[(size_t)(blockRow + mbase + r) * C1c + col] = vv;
    s += vv; s2 += vv * vv;
  }
  atomicAdd(&sum1[col], s);
  atomicAdd(&sq1[col], s2);
}

// ---------------- driver ----------------
extern "C" void kernel_launch(void* const* d_in, const int* in_sizes, int n_in,
                              void* d_out, int out_size, void* d_ws, size_t ws_size,
                              hipStream_t stream) {
  const float* points = (const float*)d_in[0];
  const float* W0 = (const float*)d_in[1];
  const float* g0 = (const float*)d_in[2];
  const float* b0 = (const float*)d_in[3];
  const float* W1 = (const float*)d_in[4];
  const float* g1 = (const float*)d_in[5];
  const float* b1 = (const float*)d_in[6];
  float* out = (float*)d_out;

  // workspace layout (fp32 elements)
  float* base  = (float*)d_ws;
  int*   vidx  = (int*)base;                            // N
  float* cnt   = base + NPTS;                           // CANVAS
  float* sx    = cnt  + CANVASc;                        // CANVAS
  float* sy    = sx   + CANVASc;                        // CANVAS
  float* sz    = sy   + CANVASc;                        // CANVAS
  float* feats = sz   + CANVASc;                        // N*16
  float* h0    = feats + (size_t)NPTS * KPAD0;          // N*64 (becomes pf0 in-place)
  float* vmax0 = h0   + (size_t)NPTS * C0c;             // CANVAS*64
  float* h1    = vmax0 + (size_t)CANVASc * C0c;         // N*128
  float* stats = h1   + (size_t)NPTS * C1c;             // 768
  float* W0p   = stats + 768;                           // 16*64 padded weights
  float *sum0 = stats,       *sq0 = stats + 64,  *a0p = stats + 128, *bs0 = stats + 192;
  float *sum1 = stats + 256, *sq1 = stats + 384, *a1p = stats + 512, *bs1 = stats + 640;

  // zero canvases / stats / output (output must realize where(occupied, ., 0))
  zero_f32<<<2048, 256, 0, stream>>>(cnt,   (size_t)4 * CANVASc);
  zero_f32<<<4096, 256, 0, stream>>>(vmax0, (size_t)CANVASc * C0c);
  zero_f32<<<4,    256, 0, stream>>>(stats, (size_t)768);
  zero_f32<<<8192, 256, 0, stream>>>(out,   (size_t)CANVASc * C1c);
  pad_w0  <<<4,    256, 0, stream>>>(W0, W0p);

  voxelize   <<<(NPTS + 255) / 256, 256, 0, stream>>>(points, vidx, cnt, sx, sy, sz);
  build_feats<<<(NPTS + 255) / 256, 256, 0, stream>>>(points, vidx, cnt, sx, sy, sz, feats);

  gemm0_wmma<<<NPTS / 32, 256, 0, stream>>>(feats, W0p, h0, sum0, sq0);
  bn_params <<<1, 64, 0, stream>>>(sum0, sq0, g0, b0, a0p, bs0, C0c);
  act_segmax<<<4096, 256, 0, stream>>>(h0, a0p, bs0, vidx, vmax0, 6, 1);

  gemm1_wmma<<<NPTS / 16, 256, 0, stream>>>(h0, vmax0, vidx, W1, h1, sum1, sq1);
  bn_params <<<1, 128, 0, stream>>>(sum1, sq1, g1, b1, a1p, bs1, C1c);
  act_segmax<<<8192, 256, 0, stream>>>(h1, a1p, bs1, vidx, out, 7, 0);
}